// GridSamplePScan_34900904247815
// MI455X (gfx1250) — compile-verified
//
#include <hip/hip_runtime.h>
#include <cstdint>
#include <cstddef>

// Problem constants (from reference): B,L,C,H,W = 2,48,32,64,64
#define BB   2
#define LLn  48
#define CCn  32
#define HHn  64
#define WWn  64
#define HWn  (HHn * WWn)      // 4096
#define TPB  256              // threads per block = pixels per slab
#define TILES (HWn / TPB)     // 16 slabs per (b,t) plane

typedef unsigned int u32t;
typedef u32t u32x4 __attribute__((ext_vector_type(4)));
typedef int  i32x4 __attribute__((ext_vector_type(4)));
typedef int  i32x8 __attribute__((ext_vector_type(8)));

__device__ __forceinline__ int clampi(int v, int lo, int hi) {
    return v < lo ? lo : (v > hi ? hi : v);
}

extern "C" __global__ __launch_bounds__(TPB)
void gsps_kernel(const float* __restrict__ flows,     // (B, L, 2, H, W)
                 const float* __restrict__ images,    // (B, L, C, H, W)
                 const float* __restrict__ decay_log, // scalar
                 float* __restrict__ out)             // (B, L, C, H, W)
{
    extern __shared__ float ldsf[];  // [2*LLn][TPB] : flow rows (l*2+axis) for this pixel slab

    const int tid  = threadIdx.x;
    const int tile = blockIdx.x & (TILES - 1);
    const int t    = (blockIdx.x / TILES) % LLn;
    const int b    = blockIdx.x / (TILES * LLn);
    const int p0   = tile * TPB;
    const int rows = 2 * (t + 1);    // flow rows actually needed (l=0..t, x&y)

    // ---------------------------------------------------------------
    // Stage flow slab into LDS with the Tensor Data Mover (one wave).
    // Tensor view of flows at base (b, l=0, ax=0, p0):
    //   dim0 = 256 contiguous floats (the pixel slab)
    //   dim1 = row index Y = 2*l + axis, row stride = H*W elements
    // LDS result: ldsf[Y*TPB + i].
    // ---------------------------------------------------------------
#if __has_builtin(__builtin_amdgcn_tensor_load_to_lds) && __has_builtin(__builtin_amdgcn_s_wait_tensorcnt)
    if (tid < 32) {  // wave 0 only (wave32): TDM is a whole-wave op, EXEC ignored
        unsigned long long ga = (unsigned long long)(uintptr_t)flows
                              + 4ull * ((unsigned long long)b * (LLn * 2) * HWn
                                        + (unsigned long long)p0);
        u32t ldsb = (u32t)(uintptr_t)(void*)&ldsf[0];  // low 32 bits = LDS byte offset

        u32x4 g0; i32x8 g1; i32x4 g2, g3; i32x8 g4;
        // D# group 0: count=1 (valid), lds_addr, global_addr[56:0], type=2
        g0.x = 1u;
        g0.y = ldsb;
        g0.z = (u32t)ga;
        g0.w = ((u32t)(ga >> 32) & 0x01FFFFFFu) | (2u << 30);
        // D# group 1
        g1[0] = (int)(2u << 16);                 // data_size=2 -> 4 bytes; no mask/pad/iterate
        g1[1] = (int)((u32t)TPB << 16);          // tensor_dim0[15:0] = 256 (bits 63:48)
        g1[2] = (int)((u32t)(2 * LLn) << 16);    // tensor_dim1[15:0] = 96  (bits 95:80)
        g1[3] = (int)((u32t)TPB << 16);          // tile_dim0 = 256 (bits 127:112)
        g1[4] = (int)rows;                       // tile_dim1 = 2*(t+1); tile_dim2 = 0
        g1[5] = HWn;                             // tensor_dim0_stride[31:0] = 4096
        g1[6] = 0;                               // stride0 hi / stride1 lo
        g1[7] = 0;                               // stride1 hi
        g2 = (i32x4){0, 0, 0, 0};                // tensor_dim2.. unused (tile_dim2=0)
        g3 = (i32x4){0, 0, 0, 0};
        g4 = (i32x8){0, 0, 0, 0, 0, 0, 0, 0};    // extra descriptor group (unused, 2D tensor)
        __builtin_amdgcn_tensor_load_to_lds(g0, g1, g2, g3, g4, 0);
        __builtin_amdgcn_s_wait_tensorcnt(0);
    }
#else
    // Fallback: cooperative copy (same LDS layout)
    for (int r = 0; r < rows; ++r)
        ldsf[r * TPB + tid] = flows[((size_t)b * (LLn * 2) + r) * HWn + p0 + tid];
#endif
    __syncthreads();

    // ---------------------------------------------------------------
    // Per-pixel setup
    // ---------------------------------------------------------------
    const int   p   = p0 + tid;
    const int   h   = p >> 6;
    const int   w   = p & (WWn - 1);
    const float sx  = 2.0f / (float)WWn;
    const float sy  = 2.0f / (float)HHn;
    const float gx0 = -1.0f + ((float)w + 0.5f) * sx;
    const float gy0 = -1.0f + ((float)h + 0.5f) * sy;
    const float decay = expf(decay_log[0]);      // decay = exp(decay_log)

    // cum_t = sum_{j<=t} flow_j at this pixel (from LDS)
    float cx = 0.0f, cy = 0.0f;
    for (int j = 0; j <= t; ++j) {
        cx += ldsf[(2 * j)     * TPB + tid];
        cy += ldsf[(2 * j + 1) * TPB + tid];
    }

    float acc[CCn];
#pragma unroll
    for (int c = 0; c < CCn; ++c) acc[c] = 0.0f;

    const float* imgB = images + (size_t)b * LLn * CCn * HWn;

    float lx = 0.0f, ly = 0.0f;
    for (int l = 0; l <= t; ++l) {
        // running cum_l
        lx += ldsf[(2 * l)     * TPB + tid];
        ly += ldsf[(2 * l + 1) * TPB + tid];

        const float wl = expf(-decay * (float)(t - l));

        float gx = gx0 + (cx - lx);
        float gy = gy0 + (cy - ly);
        // wrap gx into [-1, 1): remainder(gx+1, 2) - 1 (divisor-signed mod)
        float fx = fmodf(gx + 1.0f, 2.0f);
        fx = (fx < 0.0f) ? (fx + 2.0f) : fx;
        gx = fx - 1.0f;

        const float ix  = (gx + 1.0f) * (0.5f * (float)WWn) - 0.5f;
        const float iy  = (gy + 1.0f) * (0.5f * (float)HHn) - 0.5f;
        const float x0f = floorf(ix);
        const float y0f = floorf(iy);
        const float fxw = ix - x0f;
        const float fyw = iy - y0f;
        const int x0 = (int)x0f, y0 = (int)y0f;
        const int x1 = x0 + 1,   y1 = y0 + 1;

        const float wa = (1.0f - fxw) * (1.0f - fyw);  // (x0, y0)
        const float wb = (1.0f - fxw) * fyw;           // (x0, y1)
        const float wc = fxw * (1.0f - fyw);           // (x1, y0)
        const float wd = fxw * fyw;                    // (x1, y1)

        const float vx0 = (x0 >= 0 && x0 < WWn) ? 1.0f : 0.0f;
        const float vx1 = (x1 >= 0 && x1 < WWn) ? 1.0f : 0.0f;
        const float vy0 = (y0 >= 0 && y0 < HHn) ? 1.0f : 0.0f;
        const float vy1 = (y1 >= 0 && y1 < HHn) ? 1.0f : 0.0f;

        // fold validity and temporal weight into tap weights
        const float wA = wl * wa * vx0 * vy0;
        const float wB = wl * wb * vx0 * vy1;
        const float wC = wl * wc * vx1 * vy0;
        const float wD = wl * wd * vx1 * vy1;

        const int xc0 = clampi(x0, 0, WWn - 1);
        const int xc1 = clampi(x1, 0, WWn - 1);
        const int yc0 = clampi(y0, 0, HHn - 1);
        const int yc1 = clampi(y1, 0, HHn - 1);

        const int oa = yc0 * WWn + xc0;
        const int ob = yc1 * WWn + xc0;
        const int oc = yc0 * WWn + xc1;
        const int od = yc1 * WWn + xc1;

        const float* img = imgB + (size_t)l * CCn * HWn;  // frame (b, l), channel 0

        // hint: pull next frame's neighborhood toward this WGP (global_prefetch_b8)
        if (l < t)
            __builtin_prefetch(img + (size_t)CCn * HWn + oa, 0, 0);

        // 32 channels share the 4 taps: 4 loads + 4 FMAs per channel
#pragma unroll 8
        for (int c = 0; c < CCn; ++c) {
            const float* ic = img + (size_t)c * HWn;
            const float va = ic[oa];
            const float vb = ic[ob];
            const float vc = ic[oc];
            const float vd = ic[od];
            acc[c] = fmaf(wA, va,
                     fmaf(wB, vb,
                     fmaf(wC, vc,
                     fmaf(wD, vd, acc[c]))));
        }
    }

    // out[b, t, c, h, w]
    float* op = out + ((size_t)(b * LLn + t) * CCn) * HWn + p;
#pragma unroll 8
    for (int c = 0; c < CCn; ++c)
        op[(size_t)c * HWn] = acc[c];
}

extern "C" void kernel_launch(void* const* d_in, const int* in_sizes, int n_in,
                              void* d_out, int out_size, void* d_ws, size_t ws_size,
                              hipStream_t stream) {
    (void)in_sizes; (void)n_in; (void)d_ws; (void)ws_size; (void)out_size;
    const float* flows     = (const float*)d_in[0];  // (B, L, 2, H, W) f32
    const float* images    = (const float*)d_in[1];  // (B, L, C, H, W) f32
    const float* decay_log = (const float*)d_in[2];  // scalar f32
    float*       out       = (float*)d_out;          // (B, L, C, H, W) f32

    const dim3 grid(TILES * LLn * BB);   // 16 * 48 * 2 = 1536 workgroups
    const dim3 block(TPB);
    const size_t shmem = (size_t)(2 * LLn) * TPB * sizeof(float);  // 96 KB slab of flows

    gsps_kernel<<<grid, block, shmem, stream>>>(flows, images, decay_log, out);
}